// GCN_33449205301933
// MI455X (gfx1250) — compile-verified
//
#include <hip/hip_runtime.h>
#include <hip/hip_bf16.h>

// GCN 2-layer: out = Ahat * relu(Ahat * X * W1 + b1) * W2 + b2
// Ahat = D^-1/2 (A+I) D^-1/2.  N=100000, E=6400000, F_IN=128, H=16, F_OUT=64.

typedef __attribute__((ext_vector_type(2))) float v2f;
typedef __attribute__((ext_vector_type(8))) float v8f;

#define F_IN  128
#define HID   16
#define F_OUT 64

// ---------------- degree / norm ----------------

__global__ void k_deg_init(float* __restrict__ deg, int N) {
    int i = blockIdx.x * blockDim.x + threadIdx.x;
    if (i < N) deg[i] = 1.0f;                      // self-loop
}

__global__ void k_deg_edges(const int* __restrict__ dst, float* __restrict__ deg, int E) {
    int e = blockIdx.x * blockDim.x + threadIdx.x;
    if (e < E) atomicAdd(&deg[dst[e]], 1.0f);
}

__global__ void k_dinv(float* __restrict__ deg, int N) {
    int i = blockIdx.x * blockDim.x + threadIdx.x;
    if (i < N) deg[i] = rsqrtf(fmaxf(deg[i], 1e-12f));   // in-place: deg -> dinv
}

// ---------------- GEMM1: H1[N x 16] = X[N x 128] @ W1[128 x 16] (WMMA f32 16x16x4) ----------------

__global__ void k_gemm1(const float* __restrict__ X, const float* __restrict__ W1,
                        float* __restrict__ H1, int nTiles) {
    const int wave = blockIdx.x * (blockDim.x >> 5) + (threadIdx.x >> 5);
    if (wave >= nTiles) return;                    // wave-uniform: EXEC all ones inside
    const int lane = threadIdx.x & 31;
    const int half = lane >> 4;                    // 0: K0/K1, 1: K2/K3
    const int l16  = lane & 15;
    const int row  = wave * 16 + l16;              // M index held by this lane
    const float* __restrict__ xrow = X + (long long)row * F_IN;

    v8f c = {};
#pragma unroll
    for (int k = 0; k < F_IN; k += 4) {
        v2f a, b;
        a.x = xrow[k + 2 * half + 0];
        a.y = xrow[k + 2 * half + 1];
        b.x = W1[(k + 2 * half + 0) * HID + l16];
        b.y = W1[(k + 2 * half + 1) * HID + l16];
        c = __builtin_amdgcn_wmma_f32_16x16x4_f32(false, a, false, b,
                                                  (short)0, c, false, false);
    }
    // C/D layout: VGPR j -> M = j + 8*half, N = l16
    float* __restrict__ out = H1 + (long long)(wave * 16) * HID;
#pragma unroll
    for (int j = 0; j < 8; ++j)
        out[(j + 8 * half) * HID + l16] = c[j];
}

// ---------------- layer-1 aggregation ----------------

__global__ void k_init_agg1(const float* __restrict__ H1, const float* __restrict__ dinv,
                            float* __restrict__ agg1, int N) {
    int idx = blockIdx.x * blockDim.x + threadIdx.x;
    if (idx < N * HID) {
        int i = idx >> 4;
        float di = dinv[i];
        agg1[idx] = H1[idx] * di * di;             // self-loop message
    }
}

__global__ void k_scatter1(const int* __restrict__ src, const int* __restrict__ dst,
                           const float* __restrict__ dinv, const float* __restrict__ H1,
                           float* __restrict__ agg1, int E) {
    long long gid = (long long)blockIdx.x * blockDim.x + threadIdx.x;
    if (gid >= (long long)E * HID) return;
    int e = (int)(gid >> 4);
    int f = (int)(gid & 15);
    int s = src[e], d = dst[e];
    float w = dinv[s] * dinv[d];
    atomicAdd(&agg1[(long long)d * HID + f], H1[(long long)s * HID + f] * w);
}

__global__ void k_bias_relu(float* __restrict__ agg1, const float* __restrict__ b1, int N) {
    int idx = blockIdx.x * blockDim.x + threadIdx.x;
    if (idx < N * HID) {
        int f = idx & 15;
        agg1[idx] = fmaxf(agg1[idx] + b1[f], 0.0f);
    }
}

// ---------------- GEMM2: H3[N x 64] = H2[N x 16] @ W2[16 x 64] (WMMA f32 16x16x4) ----------------

__global__ void k_gemm2(const float* __restrict__ H2, const float* __restrict__ W2,
                        float* __restrict__ H3, int nRowTiles) {
    const int wave = blockIdx.x * (blockDim.x >> 5) + (threadIdx.x >> 5);
    const int total = nRowTiles * (F_OUT / 16);
    if (wave >= total) return;
    const int rt = wave >> 2;                      // row tile
    const int ct = wave & 3;                       // col tile (64/16)
    const int lane = threadIdx.x & 31;
    const int half = lane >> 4;
    const int l16  = lane & 15;
    const int row  = rt * 16 + l16;

    v8f c = {};
#pragma unroll
    for (int k = 0; k < HID; k += 4) {
        v2f a, b;
        a.x = H2[(long long)row * HID + k + 2 * half + 0];
        a.y = H2[(long long)row * HID + k + 2 * half + 1];
        b.x = W2[(k + 2 * half + 0) * F_OUT + ct * 16 + l16];
        b.y = W2[(k + 2 * half + 1) * F_OUT + ct * 16 + l16];
        c = __builtin_amdgcn_wmma_f32_16x16x4_f32(false, a, false, b,
                                                  (short)0, c, false, false);
    }
#pragma unroll
    for (int j = 0; j < 8; ++j)
        H3[(long long)(rt * 16 + j + 8 * half) * F_OUT + ct * 16 + l16] = c[j];
}

// ---------------- layer-2 aggregation (writes d_out) ----------------

__global__ void k_init_out(const float* __restrict__ H3, const float* __restrict__ dinv,
                           const float* __restrict__ b2, float* __restrict__ out, int N) {
    int idx = blockIdx.x * blockDim.x + threadIdx.x;
    if (idx < N * F_OUT) {
        int i = idx >> 6;
        int f = idx & 63;
        float di = dinv[i];
        out[idx] = H3[idx] * di * di + b2[f];      // self-loop + bias
    }
}

__global__ void k_scatter2(const int* __restrict__ src, const int* __restrict__ dst,
                           const float* __restrict__ dinv, const float* __restrict__ H3,
                           float* __restrict__ out, int E) {
    long long gid = (long long)blockIdx.x * blockDim.x + threadIdx.x;
    if (gid >= (long long)E * F_OUT) return;
    int e = (int)(gid >> 6);
    int f = (int)(gid & 63);
    int s = src[e], d = dst[e];
    float w = dinv[s] * dinv[d];
    atomicAdd(&out[(long long)d * F_OUT + f], H3[(long long)s * F_OUT + f] * w);
}

// ---------------- launch ----------------

extern "C" void kernel_launch(void* const* d_in, const int* in_sizes, int n_in,
                              void* d_out, int out_size, void* d_ws, size_t ws_size,
                              hipStream_t stream) {
    const float* x  = (const float*)d_in[0];
    const int*   ei = (const int*)d_in[1];       // (2, E) row-major
    const float* W1 = (const float*)d_in[2];
    const float* b1 = (const float*)d_in[3];
    const float* W2 = (const float*)d_in[4];
    const float* b2 = (const float*)d_in[5];
    float* out = (float*)d_out;

    const int N = in_sizes[0] / F_IN;
    const int E = in_sizes[1] / 2;
    const int* src = ei;
    const int* dst = ei + E;

    // workspace layout (floats): dinv[N] | H1[N*16] | agg1[N*16] | H3[N*64]
    float* ws   = (float*)d_ws;
    float* dinv = ws;
    float* H1   = dinv + N;
    float* agg1 = H1 + (long long)N * HID;
    float* H3   = agg1 + (long long)N * HID;

    const int T = 256;
    auto blk = [](long long n, int t) { return (unsigned)((n + t - 1) / t); };

    // normalization
    k_deg_init<<<blk(N, T), T, 0, stream>>>(dinv, N);
    k_deg_edges<<<blk(E, T), T, 0, stream>>>(dst, dinv, E);
    k_dinv<<<blk(N, T), T, 0, stream>>>(dinv, N);

    // layer 1
    const int nTiles1 = N / 16;                                   // 100000/16 = 6250
    k_gemm1<<<blk(nTiles1, 8), T, 0, stream>>>(x, W1, H1, nTiles1);
    k_init_agg1<<<blk((long long)N * HID, T), T, 0, stream>>>(H1, dinv, agg1, N);
    k_scatter1<<<blk((long long)E * HID, T), T, 0, stream>>>(src, dst, dinv, H1, agg1, E);
    k_bias_relu<<<blk((long long)N * HID, T), T, 0, stream>>>(agg1, b1, N);

    // layer 2
    k_gemm2<<<blk(nTiles1 * 4, 8), T, 0, stream>>>(agg1, W2, H3, nTiles1);
    k_init_out<<<blk((long long)N * F_OUT, T), T, 0, stream>>>(H3, dinv, b2, out, N);
    k_scatter2<<<blk((long long)E * F_OUT, T), T, 0, stream>>>(src, dst, dinv, H3, out, E);
}